// EPLoss_31207232372860
// MI455X (gfx1250) — compile-verified
//
#include <hip/hip_runtime.h>

// EPLoss: per-batch linear-recurrence scan over gathered pred/I columns.
// B=1024, N_T=128, N_Y=128, C=512, EOS=1.
// Strategy: async-gather (GLOBAL_LOAD_ASYNC_TO_LDS_B32, ASYNCcnt) pd/I values
// into LDS in double-buffered t-chunks; wave 0 runs the affine-map parallel
// scan with wave32 shuffles (4 j's per lane, 5 shuffle steps per t).

#define EOS_TOKEN 1

constexpr int NT  = 128;   // targets per batch
constexpr int NY  = 128;   // recurrence length
constexpr int Cc  = 512;   // classes
constexpr int TCH = 32;            // t-chunk size
constexpr int NCH = NT / TCH;      // 4 chunks

constexpr unsigned TGT_OFF = 0;                               // 128 ints
constexpr unsigned PD_OFF  = 512;                             // [2][TCH][NY] f32
constexpr unsigned IT_OFF  = PD_OFF + 2u * TCH * NY * 4u;     // [2][TCH][NY] f32
constexpr unsigned SMEM_BYTES = IT_OFF + 2u * TCH * NY * 4u;  // 66048 B

__device__ __forceinline__ void async_gather_b32(const float* gp, unsigned lds_byte) {
  unsigned long long ga = (unsigned long long)(const void*)gp;
  // vdst = per-lane LDS byte offset, vaddr = per-lane 64-bit global address
  asm volatile("global_load_async_to_lds_b32 %0, %1, off"
               :: "v"(lds_byte), "v"(ga)
               : "memory");
}

__device__ __forceinline__ void wait_async0() {
  asm volatile("s_wait_asynccnt 0" ::: "memory");
}

__global__ __launch_bounds__(128)
void eploss_kernel(const float* __restrict__ pred,
                   const float* __restrict__ R,
                   const float* __restrict__ I,
                   const int*   __restrict__ target,
                   float* __restrict__ out)
{
  __shared__ __attribute__((aligned(16))) char smem[SMEM_BYTES]; // LDS offset 0
  const int b    = blockIdx.x;
  const int tid  = threadIdx.x;
  const int wave = tid >> 5;
  const int lane = tid & 31;
  int* tgts = (int*)(smem + TGT_OFF);

  tgts[tid] = target[b * NT + tid];
  __syncthreads();

  // --- async gather of one t-chunk: wave w owns rows j in [32w, 32w+32),
  // --- lanes cover t = c*32 + lane inside row j's 2KB span (coalescer-friendly)
  auto gather_chunk = [&](int c) {
    const int buf = c & 1;
    const int t   = c * TCH + lane;
    const int tg  = tgts[t];
    const unsigned long long rowbase =
        ((unsigned long long)(b * NY + wave * 32)) * (unsigned long long)Cc;
    const float* gp = pred + rowbase + tg;
    const float* gi = I    + rowbase + tg;
    unsigned ldpd = PD_OFF + ((((unsigned)(buf * TCH + lane)) * NY + wave * 32u) << 2);
    unsigned ldit = IT_OFF + ((((unsigned)(buf * TCH + lane)) * NY + wave * 32u) << 2);
    #pragma unroll 4
    for (int jj = 0; jj < 32; ++jj) {
      async_gather_b32(gp, ldpd);
      async_gather_b32(gi, ldit);
      gp += Cc; gi += Cc; ldpd += 4; ldit += 4;
    }
  };

  gather_chunk(0);

  // --- wave-0 scan state: lane owns j = 4*lane .. 4*lane+3
  float r0v[4] = {0,0,0,0}, r1v[4] = {0,0,0,0}, r2m[4] = {1,1,1,1};
  float row[4] = {1,1,1,1};
  if (wave == 0) {
    float r2o[4];
    #pragma unroll
    for (int k = 0; k < 4; ++k) {
      const int j = 4 * lane + k;
      const float* Rp = R + ((unsigned long long)(b * NY + j)) * 3ull;
      r0v[k] = Rp[0];
      r1v[k] = Rp[1];
      r2o[k] = Rp[2];
      r2m[k] = (j > 0) ? Rp[-1] : 1.0f;   // R[b, j-1, 2]
    }
    if (lane == 31) r1v[3] = 1.0f;        // p_I[:, :, NY-1] = I_T (no R1)

    // row0[j] = (tgt0==EOS) ? 1 : prod_{k=1..j} R2[k]  (prefix product via shuffles)
    if (tgts[0] == EOS_TOKEN) {
      row[0] = row[1] = row[2] = row[3] = 1.0f;
    } else {
      const float g0 = (lane == 0) ? 1.0f : r2o[0];
      const float p0 = g0;
      const float p1 = p0 * r2o[1];
      const float p2 = p1 * r2o[2];
      const float p3 = p2 * r2o[3];
      float P = p3;
      #pragma unroll
      for (int s = 1; s < 32; s <<= 1) {
        const float Pu = __shfl_up(P, s, 32);
        if (lane >= s) P *= Pu;
      }
      float ex = __shfl_up(P, 1, 32);
      if (lane == 0) ex = 1.0f;
      row[0] = ex * p0; row[1] = ex * p1; row[2] = ex * p2; row[3] = ex * p3;
    }
  }
  wait_async0();
  __syncthreads();

  // --- main loop: scan chunk c (resident) while DMA fills chunk c+1
  for (int c = 0; c < NCH; ++c) {
    if (c + 1 < NCH) gather_chunk(c + 1);

    if (wave == 0) {
      const int buf  = c & 1;
      const int tmax = (c == NCH - 1) ? (TCH - 1) : TCH;  // 127 steps total
      for (int tt = 0; tt < tmax; ++tt) {
        const int t = c * TCH + tt;
        const unsigned base = (((unsigned)(buf * TCH + tt)) * NY + 4u * lane) << 2;
        const float4 pd = *(const float4*)(smem + PD_OFF + base);   // ds_load_b128
        const float4 Iv = *(const float4*)(smem + IT_OFF + base);
        const bool eosN = (tgts[t + 1] == EOS_TOKEN);

        const float pC0 = r0v[0] * pd.x, pC1 = r0v[1] * pd.y;
        const float pC2 = r0v[2] * pd.z, pC3 = r0v[3] * pd.w;
        const float pI0 = r1v[0] * Iv.x, pI1 = r1v[1] * Iv.y;
        const float pI2 = r1v[2] * Iv.z, pI3 = r1v[3] * Iv.w;

        const float prow = __shfl_up(row[3], 1, 32);  // row[j-1] for k=0
        const float ppC  = __shfl_up(pC3,    1, 32);  // pC[j-1]  for k=0

        const float a0 = row[0] * pI0 + ((lane == 0) ? 0.0f : prow * ppC);
        const float a1 = row[1] * pI1 + row[0] * pC0;
        const float a2 = row[2] * pI2 + row[1] * pC1;
        const float a3 = row[3] * pI3 + row[2] * pC2;
        const float d0 = (lane == 0) ? 1.0f : (eosN ? 1.0f : r2m[0]);
        const float d1 = eosN ? 1.0f : r2m[1];
        const float d2 = eosN ? 1.0f : r2m[2];
        const float d3 = eosN ? 1.0f : r2m[3];

        // compose lane's 4 affine elements: x -> A + D*x
        float A = a3 + d3 * (a2 + d2 * (a1 + d1 * a0));
        float D = d3 * d2 * d1 * d0;

        // wave32 inclusive scan of affine maps (op: a = ar + dr*al, d = dr*dl)
        #pragma unroll
        for (int s = 1; s < 32; s <<= 1) {
          const float Au = __shfl_up(A, s, 32);
          const float Du = __shfl_up(D, s, 32);
          if (lane >= s) { A += D * Au; D *= Du; }
        }
        float xin = __shfl_up(A, 1, 32);   // x at j = 4*lane - 1
        if (lane == 0) xin = 0.0f;

        const float x0 = a0 + d0 * xin;
        const float x1 = a1 + d1 * x0;
        const float x2 = a2 + d2 * x1;
        const float x3 = a3 + d3 * x2;
        row[0] = x0; row[1] = x1; row[2] = x2; row[3] = x3;
      }
    }
    wait_async0();     // own DMA for chunk c+1 complete before releasing barrier
    __syncthreads();   // LDS buffer handoff
  }

  if (wave == 0 && lane == 31) out[b] = row[3];  // row_final[:, -1]
}

extern "C" void kernel_launch(void* const* d_in, const int* in_sizes, int n_in,
                              void* d_out, int out_size, void* d_ws, size_t ws_size,
                              hipStream_t stream) {
  (void)in_sizes; (void)n_in; (void)d_ws; (void)ws_size;
  const float* pred   = (const float*)d_in[0];
  const float* R      = (const float*)d_in[1];
  const float* I      = (const float*)d_in[2];
  const int*   target = (const int*)d_in[3];
  float* out = (float*)d_out;

  dim3 grid((unsigned)out_size);  // B = 1024 blocks, one batch each
  dim3 block(128);                // 4 wave32 waves
  hipLaunchKernelGGL(eploss_kernel, grid, block, 0, stream,
                     pred, R, I, target, out);
}